// Seq2SlateLoss_64415919505714
// MI455X (gfx1250) — compile-verified
//
#include <hip/hip_runtime.h>
#include <hip/hip_bf16.h>

typedef float v2f __attribute__((ext_vector_type(2)));
typedef float v8f __attribute__((ext_vector_type(8)));

#define SEQ_L   2048
#define SEQ_B   64
#define INV_E_F 0.36787944117144233f
#define LN2_F   0.69314718055994531f

// Raw v_log_f32 (log2); ln(x) factor of ln2 is hoisted into the epilogue.
#if __has_builtin(__builtin_amdgcn_logf)
#define FAST_LOG2(x) __builtin_amdgcn_logf(x)
#else
#define FAST_LOG2(x) __log2f(x)
#endif

// Kernel 1: one wave reduces a 16-row tile of one batch matrix.
// acc1 = sum_j mask*log2(x)*y   (via WMMA f32 16x16x4 with B = ones)
// acc2 = sum_j mask*y
// Writes per-row  (-ln2*acc1)/(acc2 + 1/e) * weight[row]  to row_out.
__global__ __launch_bounds__(256) void s2s_rows_kernel(
    const float* __restrict__ logits,
    const float* __restrict__ label,
    const float* __restrict__ weight,
    float* __restrict__ row_out)
{
    __shared__ float sy[SEQ_L];

    const int tid  = threadIdx.x;
    const int wave = tid >> 5;
    const int lane = tid & 31;
    const int half = lane >> 4;   // 0: K=0,1   1: K=2,3
    const int m    = lane & 15;   // row within 16-row tile

    // 8 row-groups (16 rows each) per block; 128 groups per batch matrix,
    // so every block lives entirely inside one batch.
    const int g0 = blockIdx.x * 8;
    const int b  = g0 >> 7;

    // Stage y = label[b, :] (8 KB) into LDS once per block.
    for (int i = tid; i < SEQ_L; i += 256) sy[i] = label[b * SEQ_L + i];
    __syncthreads();

    const int g          = g0 + wave;
    const int rowInBatch = (g & 127) * 16;
    const float* rowPtr  =
        logits + ((size_t)b * SEQ_L + rowInBatch + m) * (size_t)SEQ_L;

    v8f acc1 = {};   // row sums of log2(x)*y (masked)
    v8f acc2 = {};   // row sums of y         (masked)
    const v2f ones = {1.0f, 1.0f};

    for (int k = 0; k < SEQ_L; k += 4) {
        if ((k & 127) == 0)                       // uniform branch, EXEC intact
            __builtin_prefetch(rowPtr + k + 256, 0, 1);  // global_prefetch_b8

        const int c = k + 2 * half;
        v2f x  = *(const v2f*)(rowPtr + c);       // global_load_b64, streaming
        v2f yy = *(const v2f*)(&sy[c]);           // ds_load_b64, broadcast

        v2f a1, a2;
        {
            const float xv = x.x, yv = yy.x;
            const bool  z  = (xv == 0.0f);
            const float lv = FAST_LOG2(xv);       // v_log_f32 (trans pipe)
            a1.x = z ? 0.0f : lv * yv;
            a2.x = z ? 0.0f : yv;
        }
        {
            const float xv = x.y, yv = yy.y;
            const bool  z  = (xv == 0.0f);
            const float lv = FAST_LOG2(xv);
            a1.y = z ? 0.0f : lv * yv;
            a2.y = z ? 0.0f : yv;
        }

        // D = A(16x4) x ones(4x16) + C : every output column = row sum.
        acc1 = __builtin_amdgcn_wmma_f32_16x16x4_f32(
            false, a1, false, ones, (short)0, acc1, false, false);
        acc2 = __builtin_amdgcn_wmma_f32_16x16x4_f32(
            false, a2, false, ones, (short)0, acc2, false, false);
    }

    // C/D layout: lane 0 column holds rows 0..7 in VGPRs 0..7,
    // lane 16 column holds rows 8..15.
    if (m == 0) {
        const int mbase = half * 8;
#pragma unroll
        for (int j = 0; j < 8; ++j) {
            const int   row = rowInBatch + mbase + j;
            const float num = -LN2_F * acc1[j];   // fold ln2 here, once
            const float den = acc2[j] + INV_E_F;
            row_out[(size_t)b * SEQ_L + row] = num / den * weight[row];
        }
    }
}

// Kernel 2: deterministic fixed-order reduction of B*L row values -> scalar.
__global__ __launch_bounds__(1024) void s2s_reduce_kernel(
    const float* __restrict__ row_out, float* __restrict__ out)
{
    __shared__ float sbuf[1024];
    const int tid = threadIdx.x;

    float acc = 0.0f;
    for (int i = tid; i < SEQ_B * SEQ_L; i += 1024) acc += row_out[i];
    sbuf[tid] = acc;
    __syncthreads();

    for (int s = 512; s > 0; s >>= 1) {
        if (tid < s) sbuf[tid] += sbuf[tid + s];
        __syncthreads();
    }
    if (tid == 0) out[0] = sbuf[0] * (1.0f / SEQ_B);
}

extern "C" void kernel_launch(void* const* d_in, const int* in_sizes, int n_in,
                              void* d_out, int out_size, void* d_ws, size_t ws_size,
                              hipStream_t stream) {
    (void)in_sizes; (void)n_in; (void)out_size; (void)ws_size;

    const float* logits = (const float*)d_in[0];   // (64, 1, 2048, 2048) f32
    const float* label  = (const float*)d_in[1];   // (64, 2048) f32
    const float* weight = (const float*)d_in[2];   // (2048,) f32
    float*       out    = (float*)d_out;           // scalar f32
    float*       rows   = (float*)d_ws;            // B*L floats = 512 KB scratch

    // 131072 rows / 16 rows per wave = 8192 waves = 1024 blocks x 8 waves.
    s2s_rows_kernel<<<1024, 256, 0, stream>>>(logits, label, weight, rows);
    s2s_reduce_kernel<<<1, 1024, 0, stream>>>(rows, out);
}